// ClinicalSafetyLoss_84146999263816
// MI455X (gfx1250) — compile-verified
//
#include <hip/hip_runtime.h>

// ---------------------------------------------------------------------------
// ClinicalSafetyLoss for MI455X (gfx1250, wave32).
// Kernel 1: memory-bound streaming reduction (b128 loads, 4 samples/thread/iter),
//           wave32 shuffle + LDS block reduce -> one partial row [ce,sp,nc,ms] per block.
// Kernel 2: single wave; per-lane strided accumulation over partial rows, then ONE
//           V_WMMA_F32_16X16X4_F32 with an all-ones B matrix performs the final
//           32-lane cross-lane reduction (D[m][n] = sum_k A[m][k]).
// ---------------------------------------------------------------------------

typedef __attribute__((ext_vector_type(2))) float v2f;
typedef __attribute__((ext_vector_type(8))) float v8f;

#define NTHREADS 256
#define WAVES_PER_BLOCK (NTHREADS / 32)

// Penalty matrix packed 5 bits per entry, index = t*3 + pred:
// [0,1,2, 5,0,1, 20,10,0]
__device__ __constant__ const unsigned long long kPenLut =
    (0ull << 0) | (1ull << 5) | (2ull << 10) |
    (5ull << 15) | (0ull << 20) | (1ull << 25) |
    (20ull << 30) | (10ull << 35) | (0ull << 40);

__device__ __forceinline__ void accum_sample(float x0, float x1, float x2, int t,
                                             float& ce, float& sp, float& nc, float& ms) {
    // argmax with first-occurrence tie-break (strict >)
    int pred = 0;
    float mx = x0;
    if (x1 > mx) { pred = 1; mx = x1; }
    if (x2 > mx) { pred = 2; mx = x2; }
    // log-sum-exp (numerically shifted), v_exp_f32 / v_log_f32 transcendentals
    float s = __expf(x0 - mx) + __expf(x1 - mx) + __expf(x2 - mx);
    float lse = mx + __logf(s);
    float xt = (t == 0) ? x0 : ((t == 1) ? x1 : x2);
    ce += lse - xt;
    // penalty gather via packed-immediate shift (no scratch-array indexing)
    int idx = t * 3 + pred;
    sp += (float)((kPenLut >> (5 * idx)) & 31ull);
    nc += (t == 2) ? 1.0f : 0.0f;
    ms += (t == 2 && pred != 2) ? 1.0f : 0.0f;
}

__global__ void clinical_loss_partial_kernel(const float* __restrict__ logits,
                                             const int* __restrict__ targets,
                                             float* __restrict__ partials,
                                             int ngroups) {
    float ce = 0.f, sp = 0.f, nc = 0.f, ms = 0.f;

    const int stride = gridDim.x * blockDim.x;
    for (int g = blockIdx.x * blockDim.x + threadIdx.x; g < ngroups; g += stride) {
        // 4 samples = 12 floats = 3 x b128 loads (48-byte chunks, 16-B aligned)
        const float4* p = reinterpret_cast<const float4*>(logits + (size_t)g * 12);
        float4 a = p[0];
        float4 b = p[1];
        float4 c = p[2];
        int4 t = reinterpret_cast<const int4*>(targets)[g];

        accum_sample(a.x, a.y, a.z, t.x, ce, sp, nc, ms);
        accum_sample(a.w, b.x, b.y, t.y, ce, sp, nc, ms);
        accum_sample(b.z, b.w, c.x, t.z, ce, sp, nc, ms);
        accum_sample(c.y, c.z, c.w, t.w, ce, sp, nc, ms);
    }

    // wave32 butterfly reduction
    for (int off = 16; off; off >>= 1) {
        ce += __shfl_xor(ce, off, 32);
        sp += __shfl_xor(sp, off, 32);
        nc += __shfl_xor(nc, off, 32);
        ms += __shfl_xor(ms, off, 32);
    }

    __shared__ float smem[WAVES_PER_BLOCK][4];
    const int lane = threadIdx.x & 31;
    const int wave = threadIdx.x >> 5;
    if (lane == 0) {
        smem[wave][0] = ce;
        smem[wave][1] = sp;
        smem[wave][2] = nc;
        smem[wave][3] = ms;
    }
    __syncthreads();
    if (threadIdx.x < 4) {
        float v = 0.f;
        #pragma unroll
        for (int w = 0; w < WAVES_PER_BLOCK; ++w) v += smem[w][threadIdx.x];
        partials[blockIdx.x * 4 + threadIdx.x] = v;  // no atomics -> deterministic
    }
}

// Single-wave finalize. Partial rows: [nrows][4] = {ce, sp, nc, ms}.
// Lane l: M = l&15 (component, only 0..3 live), K-pair = (l>>4)? {2,3} : {0,1}.
// One f32 WMMA with B = ones computes D[m][*] = sum of the 4 A values held by lanes m and m+16.
__global__ void clinical_loss_final_kernel(const float* __restrict__ partials,
                                           float* __restrict__ out,
                                           int nrows, float inv_batch) {
    const int lane = threadIdx.x;     // 0..31
    const int m = lane & 15;
    const int khi = lane >> 4;        // 0 -> rows +0,+1 ; 1 -> rows +2,+3

    // Mask-multiply instead of divergent control flow: EXEC stays all-ones,
    // as required by the WMMA restriction.
    const float w = (m < 4) ? 1.0f : 0.0f;
    const int mc = m & 3;

    float acc0 = 0.f, acc1 = 0.f;
    for (int base = 0; base < nrows; base += 4) {
        acc0 += w * partials[(base + 2 * khi) * 4 + mc];
        acc1 += w * partials[(base + 2 * khi + 1) * 4 + mc];
    }

    v2f A;  A[0] = acc0; A[1] = acc1;
    v2f Bv; Bv[0] = 1.0f; Bv[1] = 1.0f;   // all-ones 4x16 B matrix
    v8f C = {};
    // D[m][n] = sum_k A[m][k]  -> per-component totals replicated across N
    C = __builtin_amdgcn_wmma_f32_16x16x4_f32(
        /*neg_a=*/false, A, /*neg_b=*/false, Bv,
        /*c_mod=*/(short)0, C, /*reuse_a=*/false, /*reuse_b=*/false);

    if (lane == 0) {
        // 16x16 f32 D layout: lane 0 VGPR m holds D[m][0] = component-m total
        float ce_tot = C[0];
        float sp_tot = C[1];
        float nc_tot = C[2];
        float ms_tot = C[3];
        float ce_mean = ce_tot * inv_batch;
        float sp_mean = sp_tot * inv_batch;
        float crit = (nc_tot > 0.f) ? (ms_tot / fmaxf(nc_tot, 1.0f)) * 20.0f : 0.0f;
        out[0] = ce_mean + 0.3f * sp_mean + 0.5f * crit;
    }
}

extern "C" void kernel_launch(void* const* d_in, const int* in_sizes, int n_in,
                              void* d_out, int out_size, void* d_ws, size_t ws_size,
                              hipStream_t stream) {
    (void)n_in; (void)out_size;
    const float* logits = (const float*)d_in[0];   // [B,3] f32
    const int* targets = (const int*)d_in[1];      // [B] int32
    float* out = (float*)d_out;                    // scalar f32
    float* partials = (float*)d_ws;                // [nblk][4] f32

    const int B = in_sizes[1];
    const int ngroups = B / 4;                     // B = 4194304, divisible by 4

    // One partial row (16 B) per block; cap by workspace, keep multiple of 4.
    int nblk = 2048;
    int maxblk = (int)(ws_size / (4 * sizeof(float)));
    if (nblk > maxblk) nblk = maxblk & ~3;
    if (nblk < 4) nblk = 4;

    clinical_loss_partial_kernel<<<nblk, NTHREADS, 0, stream>>>(logits, targets, partials, ngroups);
    clinical_loss_final_kernel<<<1, 32, 0, stream>>>(partials, out, nblk, 1.0f / (float)B);
}